// ManifoldSKI_38182259262064
// MI455X (gfx1250) — compile-verified
//
#include <hip/hip_runtime.h>
#include <math.h>

// ManifoldSKI on MI455X (gfx1250): fp32 WMMA 16x16x4 for all GEMM-shaped work.
// B=512, D=128, K=11, V=50257. Fixed 40 fixed-point iterations (deterministic).

#define B_SZ   512
#define D_SZ   128
#define K_OPS  11
#define V_SZ   50257
#define N_ITER 40
#define BETA_C 5.0f
#define LDW    132   // padded LDS row stride (132 % 64 == 4 -> conflict-free A loads)

typedef float v2f __attribute__((ext_vector_type(2)));
typedef float v8f __attribute__((ext_vector_type(8)));

__device__ __forceinline__ v8f wmma4(v2f a, v2f b, v8f c) {
    // D = A(16x4 f32) * B(4x16 f32) + C(16x16 f32)
    return __builtin_amdgcn_wmma_f32_16x16x4_f32(
        /*neg_a=*/false, a, /*neg_b=*/false, b,
        /*c_mod=*/(short)0, c, /*reuse_a=*/false, /*reuse_b=*/false);
}

// A-fragment (16x4 tile at k-offset kb) from a row-major [16][LDW] LDS tile.
// Layout: lanes 0-15 -> M=lane, K = kb+{0,1}; lanes 16-31 -> M=lane-16, K = kb+{2,3}.
__device__ __forceinline__ v2f lds_afrag(const float* tile, int kb, int lane) {
    int m  = lane & 15;
    int kh = (lane >> 4) << 1;
    const float* p = tile + m * LDW + kb + kh;
    v2f a; a.x = p[0]; a.y = p[1];
    return a;
}

// B-fragment for Y = X @ Wt^T: B[k,n] = Wt[n0+n][kb+k], Wt row-major with leading dim ld.
// Layout mirrors A: lanes 0-15 -> N=lane, K = kb+{0,1}; lanes 16-31 -> N=lane-16, K = kb+{2,3}.
__device__ __forceinline__ v2f glb_bfrag(const float* mat, int ld, int n0, int kb, int lane) {
    int n  = lane & 15;
    int kh = (lane >> 4) << 1;
    const float* p = mat + (size_t)(n0 + n) * ld + kb + kh;
    float2 t = *reinterpret_cast<const float2*>(p);
    v2f b; b.x = t.x; b.y = t.y;
    return b;
}

// ---------------- Kernel 1: h_ctx, f_emb gather, softmax address weights ----
__global__ __launch_bounds__(D_SZ)
void ski_setup_kernel(const float* __restrict__ depth, const float* __restrict__ complexity,
                      const float* __restrict__ op_emb, const float* __restrict__ addr_m,
                      const float* __restrict__ dep_w, const float* __restrict__ dep_b,
                      const float* __restrict__ comp_w, const float* __restrict__ comp_b,
                      const int* __restrict__ op_ids,
                      float* __restrict__ h_ctx, float* __restrict__ f_emb,
                      float* __restrict__ a_out) {
    __shared__ float sh[D_SZ];
    __shared__ float sl[K_OPS];
    const int b = blockIdx.x;
    const int t = threadIdx.x;
    const float dv = depth[b], cv = complexity[b];
    float h = dv * dep_w[t] + dep_b[t] + cv * comp_w[t] + comp_b[t];
    h_ctx[b * D_SZ + t] = h;
    f_emb[b * D_SZ + t] = op_emb[op_ids[b] * D_SZ + t];
    sh[t] = h;
    __syncthreads();
    if (t < K_OPS) {
        const float* ar = addr_m + t * D_SZ;
        float dot = 0.f;
        #pragma unroll 8
        for (int j = 0; j < D_SZ; ++j) dot += sh[j] * ar[j];
        sl[t] = BETA_C * dot;
    }
    __syncthreads();
    if (t == 0) {
        float mx = sl[0];
        for (int k = 1; k < K_OPS; ++k) mx = fmaxf(mx, sl[k]);
        float e[K_OPS]; float s = 0.f;
        for (int k = 0; k < K_OPS; ++k) { e[k] = expf(sl[k] - mx); s += e[k]; }
        float inv = 1.f / s;
        for (int k = 0; k < K_OPS; ++k) a_out[b * 12 + k] = e[k] * inv;
    }
}

// ---------------- Kernel 2: persistent fixed-point solver ------------------
// One block per 16 batch rows; 8 waves, each owning a 16-column output tile.
__global__ __launch_bounds__(256)
void ski_fixed_point_kernel(const float* __restrict__ z_init, const float* __restrict__ W,
                            const float* __restrict__ U, const float* __restrict__ Vm,
                            const float* __restrict__ sw1, const float* __restrict__ sb1,
                            const float* __restrict__ sw2, const float* __restrict__ sb2,
                            const float* __restrict__ h_ctx, const float* __restrict__ f_emb,
                            const float* __restrict__ a_in, float* __restrict__ z_out) {
    __shared__ float Zs[16 * LDW];   // current z tile
    __shared__ float Zns[16 * LDW];  // z_new tile (prologue: f_emb tile)
    __shared__ float Hs[16 * LDW];   // stabilizer hidden (prologue: h_ctx tile)
    __shared__ float Cs[16 * LDW];   // loop-invariant term
    __shared__ float As[16 * 12];    // softmax op weights per row

    const int tid   = threadIdx.x;
    const int lane  = tid & 31;
    const int wave  = tid >> 5;           // 0..7
    const int rb    = blockIdx.x * 16;    // base batch row
    const int n0    = wave * 16;          // this wave's column tile
    const int nlane = lane & 15;
    const int mhi   = (lane >> 4) << 3;   // 0 or 8

    for (int i = tid; i < 16 * D_SZ; i += 256) {
        int r = i >> 7, c = i & 127;
        Zs [r * LDW + c] = z_init[(rb + r) * D_SZ + c];
        Hs [r * LDW + c] = h_ctx [(rb + r) * D_SZ + c];
        Zns[r * LDW + c] = f_emb [(rb + r) * D_SZ + c];
    }
    for (int i = tid; i < 16 * 12; i += 256) {
        int r = i / 12, k = i - r * 12;
        As[i] = (k < K_OPS) ? a_in[(rb + r) * 12 + k] : 0.f;
    }
    __syncthreads();

    // Loop-invariant term: Cs = sum_k a_k .* (H @ U_k^T + F @ V_k^T)
    {
        v8f acc; 
        #pragma unroll
        for (int v = 0; v < 8; ++v) acc[v] = 0.f;
        for (int k = 0; k < K_OPS; ++k) {
            v8f y;
            #pragma unroll
            for (int v = 0; v < 8; ++v) y[v] = 0.f;
            const float* Uk = U  + (size_t)k * D_SZ * D_SZ;
            const float* Vk = Vm + (size_t)k * D_SZ * D_SZ;
            for (int kb = 0; kb < D_SZ; kb += 4)
                y = wmma4(lds_afrag(Hs, kb, lane), glb_bfrag(Uk, D_SZ, n0, kb, lane), y);
            for (int kb = 0; kb < D_SZ; kb += 4)
                y = wmma4(lds_afrag(Zns, kb, lane), glb_bfrag(Vk, D_SZ, n0, kb, lane), y);
            #pragma unroll
            for (int v = 0; v < 8; ++v) acc[v] += As[(v + mhi) * 12 + k] * y[v];
        }
        #pragma unroll
        for (int v = 0; v < 8; ++v) Cs[(v + mhi) * LDW + n0 + nlane] = acc[v];
    }

    const float bias1 = sb1[n0 + nlane];
    const float bias2 = sb2[n0 + nlane];

    for (int it = 0; it < N_ITER; ++it) {
        __syncthreads();
        // Phase 1: pre = Cs + sum_k a_k .* (Z @ W_k^T); z_new = tanh(pre)
        v8f acc;
        #pragma unroll
        for (int v = 0; v < 8; ++v) acc[v] = Cs[(v + mhi) * LDW + n0 + nlane];
        for (int k = 0; k < K_OPS; ++k) {
            v8f y;
            #pragma unroll
            for (int v = 0; v < 8; ++v) y[v] = 0.f;
            const float* Wk = W + (size_t)k * D_SZ * D_SZ;
            for (int kb = 0; kb < D_SZ; kb += 4)
                y = wmma4(lds_afrag(Zs, kb, lane), glb_bfrag(Wk, D_SZ, n0, kb, lane), y);
            #pragma unroll
            for (int v = 0; v < 8; ++v) acc[v] += As[(v + mhi) * 12 + k] * y[v];
        }
        #pragma unroll
        for (int v = 0; v < 8; ++v)
            Zns[(v + mhi) * LDW + n0 + nlane] = tanhf(acc[v]);
        __syncthreads();

        // Phase 2: Hs = tanh([Z | Zn] @ sw1^T + sb1)   (K = 256)
        v8f y1;
        #pragma unroll
        for (int v = 0; v < 8; ++v) y1[v] = 0.f;
        for (int kb = 0; kb < D_SZ; kb += 4)
            y1 = wmma4(lds_afrag(Zs, kb, lane), glb_bfrag(sw1, 2 * D_SZ, n0, kb, lane), y1);
        for (int kb = 0; kb < D_SZ; kb += 4)
            y1 = wmma4(lds_afrag(Zns, kb, lane), glb_bfrag(sw1, 2 * D_SZ, n0, D_SZ + kb, lane), y1);
        #pragma unroll
        for (int v = 0; v < 8; ++v)
            Hs[(v + mhi) * LDW + n0 + nlane] = tanhf(y1[v] + bias1);
        __syncthreads();

        // Phase 3: gate = sigmoid(Hs @ sw2^T + sb2); z = gate*z_new + (1-gate)*z
        v8f y2;
        #pragma unroll
        for (int v = 0; v < 8; ++v) y2[v] = 0.f;
        for (int kb = 0; kb < D_SZ; kb += 4)
            y2 = wmma4(lds_afrag(Hs, kb, lane), glb_bfrag(sw2, D_SZ, n0, kb, lane), y2);
        #pragma unroll
        for (int v = 0; v < 8; ++v) {
            float g = 1.f / (1.f + expf(-(y2[v] + bias2)));
            int idx = (v + mhi) * LDW + n0 + nlane;
            Zs[idx] = g * Zns[idx] + (1.f - g) * Zs[idx];
        }
    }
    __syncthreads();
    for (int i = tid; i < 16 * D_SZ; i += 256) {
        int r = i >> 7, c = i & 127;
        z_out[(rb + r) * D_SZ + c] = Zs[r * LDW + c];
    }
}

// ---------------- Kernel 3: decoder GEMM  logits = Z @ dec_w^T + dec_b ------
// grid.x covers N (8 waves x 16 cols per block), grid.y covers M (16 rows).
__global__ __launch_bounds__(256)
void ski_decode_kernel(const float* __restrict__ z, const float* __restrict__ dec_w,
                       const float* __restrict__ dec_b, float* __restrict__ out) {
    __shared__ float Zt[16 * LDW];
    const int tid  = threadIdx.x;
    const int lane = tid & 31;
    const int wave = tid >> 5;
    const int rb   = blockIdx.y * 16;

    for (int i = tid; i < 16 * D_SZ; i += 256) {
        int r = i >> 7, c = i & 127;
        Zt[r * LDW + c] = z[(rb + r) * D_SZ + c];
    }
    __syncthreads();

    const int n0g = (blockIdx.x * 8 + wave) * 16;
    if (n0g >= V_SZ) return;                 // wave-uniform: EXEC stays all-1 for WMMA

    const int nlane = lane & 15;
    const int mhi   = (lane >> 4) << 3;
    const int kh    = (lane >> 4) << 1;
    const int col   = n0g + nlane;
    const int colc  = (col < V_SZ) ? col : (V_SZ - 1);   // clamped load for edge tile

    v8f y;
    #pragma unroll
    for (int v = 0; v < 8; ++v) y[v] = 0.f;
    const float* wrow = dec_w + (size_t)colc * D_SZ;
    #pragma unroll 8
    for (int kb = 0; kb < D_SZ; kb += 4) {
        v2f a = lds_afrag(Zt, kb, lane);
        float2 t2 = *reinterpret_cast<const float2*>(wrow + kb + kh);
        v2f b; b.x = t2.x; b.y = t2.y;
        y = wmma4(a, b, y);
    }
    const float bias = dec_b[colc];
    if (col < V_SZ) {
        #pragma unroll
        for (int v = 0; v < 8; ++v)
            out[(size_t)(rb + v + mhi) * V_SZ + col] = y[v] + bias;
    }
}

// ---------------------------------------------------------------------------
extern "C" void kernel_launch(void* const* d_in, const int* in_sizes, int n_in,
                              void* d_out, int out_size, void* d_ws, size_t ws_size,
                              hipStream_t stream) {
    (void)in_sizes; (void)n_in; (void)out_size; (void)ws_size;
    const float* depth      = (const float*)d_in[0];
    const float* complexity = (const float*)d_in[1];
    const float* z_init     = (const float*)d_in[2];
    const float* op_emb     = (const float*)d_in[3];
    const float* addr_m     = (const float*)d_in[4];
    const float* W          = (const float*)d_in[5];
    const float* U          = (const float*)d_in[6];
    const float* Vm         = (const float*)d_in[7];
    const float* dep_w      = (const float*)d_in[8];
    const float* dep_b      = (const float*)d_in[9];
    const float* comp_w     = (const float*)d_in[10];
    const float* comp_b     = (const float*)d_in[11];
    const float* sw1        = (const float*)d_in[12];
    const float* sb1        = (const float*)d_in[13];
    const float* sw2        = (const float*)d_in[14];
    const float* sb2        = (const float*)d_in[15];
    const float* dec_w      = (const float*)d_in[16];
    const float* dec_b      = (const float*)d_in[17];
    const int*   op_ids     = (const int*)d_in[18];
    float* out = (float*)d_out;

    float* ws    = (float*)d_ws;
    float* h_ctx = ws;                       // B*D
    float* f_emb = h_ctx + B_SZ * D_SZ;      // B*D
    float* a_ws  = f_emb + B_SZ * D_SZ;      // B*12
    float* z_ws  = a_ws + B_SZ * 12;         // B*D

    ski_setup_kernel<<<B_SZ, D_SZ, 0, stream>>>(
        depth, complexity, op_emb, addr_m, dep_w, dep_b, comp_w, comp_b, op_ids,
        h_ctx, f_emb, a_ws);

    ski_fixed_point_kernel<<<B_SZ / 16, 256, 0, stream>>>(
        z_init, W, U, Vm, sw1, sb1, sw2, sb2, h_ctx, f_emb, a_ws, z_ws);

    dim3 gdec((V_SZ + 8 * 16 - 1) / (8 * 16), B_SZ / 16);
    ski_decode_kernel<<<gdec, 256, 0, stream>>>(z_ws, dec_w, dec_b, out);
}